// LlamaBlock_89601607729570
// MI455X (gfx1250) — compile-verified
//
#include <hip/hip_runtime.h>
#include <math.h>

#define HIDDEN 4096
#define NHEADS 32
#define HDIM   128
#define INTERM 11008
#define KVLEN  4096
#define SPLITS 8
#define ATTN_SCALE 0.08838834764831845f /* 1/sqrt(128) */

typedef __attribute__((ext_vector_type(2))) float v2f;
typedef __attribute__((ext_vector_type(8))) float v8f;

// D(16x16) = A(16x4) x B(4x16) + C.  We use only row 0 of A (the activation
// chunk) so D row 0 = 16 output columns' partial dots.  Layouts per ISA 7.12.2.
__device__ __forceinline__ v8f wmma_f32_16x16x4(v2f a, v2f b, v8f c) {
#if __has_builtin(__builtin_amdgcn_wmma_f32_16x16x4_f32)
  return __builtin_amdgcn_wmma_f32_16x16x4_f32(false, a, false, b, (short)0, c,
                                               false, false);
#else
  asm volatile("v_wmma_f32_16x16x4_f32 %0, %1, %2, %0"
               : "+v"(c) : "v"(a), "v"(b));
  return c;
#endif
}

// ---------------------------------------------------------------- rmsnorm ---
__global__ void k_rmsnorm(const float* __restrict__ x, const float* __restrict__ w,
                          float* __restrict__ o, int n) {
  __shared__ float red[256];
  int tid = threadIdx.x;
  float s = 0.f;
  for (int i = tid; i < n; i += 256) { float v = x[i]; s += v * v; }
  red[tid] = s;
  __syncthreads();
  for (int off = 128; off > 0; off >>= 1) {
    if (tid < off) red[tid] += red[tid + off];
    __syncthreads();
  }
  float rs = rsqrtf(red[0] / (float)n + 1e-6f);
  for (int i = tid; i < n; i += 256) o[i] = x[i] * rs * w[i];
}

// ------------------------------------------------------------ WMMA GEMV ----
// out[j] = epilogue( sum_k xin[k] * W[k][j] ),  W row-major [K, N].
// grid.x = N/16 blocks, 256 threads = 8 waves.  Each wave owns the block's 16
// columns over a K/8 slice (split-K merged in LDS).  Inner step streams a
// 4-row x 16-col weight tile (256 B) into the B operand (coalesced 64 B
// requests; halves of 128 B lines are picked up by the neighbor block via the
// 192 MB L2) and issues one v_wmma_f32_16x16x4_f32.
// mode: 0 = store, 1 = silu, 2 = res[j] + val
__global__ void k_gemv_wmma(const float* __restrict__ W, const float* __restrict__ xin,
                            const float* __restrict__ res, float* __restrict__ out,
                            int K, int N, int mode) {
  extern __shared__ float xs[];
  __shared__ float red[8][16];
  int tid = threadIdx.x;
  for (int i = tid * 4; i < K; i += 1024)
    *(float4*)(xs + i) = *(const float4*)(xin + i);
  __syncthreads();

  int wave = tid >> 5;
  int lane = tid & 31;
  int ln = lane & 15;
  int hi = lane >> 4;                 // 0: K pair {0,1}, 1: K pair {2,3}
  float msk = (ln == 0) ? 1.f : 0.f;  // A has only row m==0 non-zero
  int j0 = blockIdx.x * 16;
  int kchunk = K / 8;                 // multiple of 4 (4096/8, 11008/8)
  int kbeg = wave * kchunk;
  size_t strideN = (size_t)N;
  const float* wp = W + (size_t)(kbeg + 2 * hi) * strideN + (j0 + ln);

  v8f acc = {0.f, 0.f, 0.f, 0.f, 0.f, 0.f, 0.f, 0.f};
  for (int k2 = 0; k2 < kchunk; k2 += 4) {
    int kk = kbeg + k2 + 2 * hi;
    v2f a, b;
    a.x = xs[kk] * msk;       // A[0][2*hi+0]
    a.y = xs[kk + 1] * msk;   // A[0][2*hi+1]
    b.x = wp[0];              // B[2*hi+0][ln] = W[kk][j0+ln]
    b.y = wp[strideN];        // B[2*hi+1][ln]
    wp += strideN * 4;
    acc = wmma_f32_16x16x4(a, b, acc);
  }
  // D row 0 lives in acc[0], lanes 0..15
  if (lane < 16) red[wave][ln] = acc[0];
  __syncthreads();
  if (tid < 16) {
    float s = 0.f;
#pragma unroll
    for (int w = 0; w < 8; ++w) s += red[w][tid];
    int j = j0 + tid;
    if (mode == 1) s = s / (1.f + __expf(-s));      // silu
    else if (mode == 2) s = res[j] + s;             // residual add
    out[j] = s;
  }
}

// ------------------------------------------------------------------ RoPE ---
__global__ void k_rope(float* __restrict__ q, float* __restrict__ k,
                       const float* __restrict__ sinc, const float* __restrict__ cosc,
                       const int* __restrict__ pos) {
  int i = blockIdx.x * blockDim.x + threadIdx.x;  // 0..2047
  int head = i >> 6, d = i & 63;
  int p = pos[0];
  float sn = sinc[(size_t)p * 64 + d];
  float cs = cosc[(size_t)p * 64 + d];
  int b = head * HDIM + d;
  float q1 = q[b], q2 = q[b + 64];
  q[b] = q1 * cs - q2 * sn;
  q[b + 64] = q2 * cs + q1 * sn;
  float k1 = k[b], k2v = k[b + 64];
  k[b] = k1 * cs - k2v * sn;
  k[b + 64] = k2v * cs + k1 * sn;
}

// ----------------------------------------------- flash-decode partials -----
// grid = (NHEADS, SPLITS), block = 256 (8 waves). Each wave walks 64
// positions; per position one coalesced 512 B K row + 512 B V row (float4 per
// lane), online softmax with wave-wide shfl_xor reduction.
__global__ void k_attn_partial(const float* __restrict__ q, const float* __restrict__ kc,
                               const float* __restrict__ vc,
                               float* __restrict__ po, float* __restrict__ pm,
                               float* __restrict__ pl) {
  int h = blockIdx.x, sp = blockIdx.y;
  int wave = threadIdx.x >> 5, lane = threadIdx.x & 31;
  float4 qv = *(const float4*)(q + h * HDIM + lane * 4);
  const int per_split = KVLEN / SPLITS;       // 512
  const int per_wave = per_split / 8;         // 64
  int tbeg = sp * per_split + wave * per_wave;
  int tend = tbeg + per_wave;

  float m = -1e30f, l = 0.f;
  float4 oa = {0.f, 0.f, 0.f, 0.f};
  for (int t = tbeg; t < tend; ++t) {
    size_t base = ((size_t)t * NHEADS + h) * HDIM + lane * 4;
    float4 kv = *(const float4*)(kc + base);
    float s = qv.x * kv.x + qv.y * kv.y + qv.z * kv.z + qv.w * kv.w;
    for (int off = 16; off > 0; off >>= 1) s += __shfl_xor(s, off, 32);
    s *= ATTN_SCALE;
    float mn = fmaxf(m, s);
    float corr = __expf(m - mn);
    float p = __expf(s - mn);
    float4 vv = *(const float4*)(vc + base);
    oa.x = oa.x * corr + p * vv.x;
    oa.y = oa.y * corr + p * vv.y;
    oa.z = oa.z * corr + p * vv.z;
    oa.w = oa.w * corr + p * vv.w;
    l = l * corr + p;
    m = mn;
  }
  __shared__ float sm[8], sl[8];
  __shared__ float so[8][HDIM];
  if (lane == 0) { sm[wave] = m; sl[wave] = l; }
  so[wave][lane * 4 + 0] = oa.x;
  so[wave][lane * 4 + 1] = oa.y;
  so[wave][lane * 4 + 2] = oa.z;
  so[wave][lane * 4 + 3] = oa.w;
  __syncthreads();
  if (threadIdx.x < HDIM) {
    int d = threadIdx.x;
    float M = sm[0];
    for (int w = 1; w < 8; ++w) M = fmaxf(M, sm[w]);
    float L = 0.f, O = 0.f;
    for (int w = 0; w < 8; ++w) {
      float c = __expf(sm[w] - M);
      L += c * sl[w];
      O += c * so[w][d];
    }
    int idx = h * SPLITS + sp;
    po[(size_t)idx * HDIM + d] = O;
    if (d == 0) { pm[idx] = M; pl[idx] = L; }
  }
}

// ------------------------------------- combine splits + current token ------
__global__ void k_attn_combine(const float* __restrict__ q, const float* __restrict__ kn,
                               const float* __restrict__ vn,
                               const float* __restrict__ po, const float* __restrict__ pm,
                               const float* __restrict__ pl, float* __restrict__ o) {
  int h = blockIdx.x, d = threadIdx.x;  // 128 threads
  __shared__ float red[HDIM];
  red[d] = q[h * HDIM + d] * kn[h * HDIM + d];
  __syncthreads();
  for (int off = 64; off > 0; off >>= 1) {
    if (d < off) red[d] += red[d + off];
    __syncthreads();
  }
  float snew = red[0] * ATTN_SCALE;   // score of the appended token
  float M = snew;
  for (int s = 0; s < SPLITS; ++s) M = fmaxf(M, pm[h * SPLITS + s]);
  float cn = __expf(snew - M);
  float L = cn;                        // new token: l contribution = p * 1
  float O = cn * vn[h * HDIM + d];
  for (int s = 0; s < SPLITS; ++s) {
    int idx = h * SPLITS + s;
    float c = __expf(pm[idx] - M);
    L += c * pl[idx];
    O += c * po[(size_t)idx * HDIM + d];
  }
  o[h * HDIM + d] = O / L;
}

// --------------------------------------------------------------- launch ----
extern "C" void kernel_launch(void* const* d_in, const int* in_sizes, int n_in,
                              void* d_out, int out_size, void* d_ws, size_t ws_size,
                              hipStream_t stream) {
  (void)in_sizes; (void)n_in; (void)out_size; (void)ws_size;
  const float* x        = (const float*)d_in[0];
  const float* w_q      = (const float*)d_in[1];
  const float* w_k      = (const float*)d_in[2];
  const float* w_v      = (const float*)d_in[3];
  const float* w_o      = (const float*)d_in[4];
  const float* attn_nrm = (const float*)d_in[5];
  const float* w_ff1    = (const float*)d_in[6];
  const float* w_ff2    = (const float*)d_in[7];
  const float* ffn_nrm  = (const float*)d_in[8];
  const float* sin_c    = (const float*)d_in[9];
  const float* cos_c    = (const float*)d_in[10];
  const float* k_cache  = (const float*)d_in[11];
  const float* v_cache  = (const float*)d_in[12];
  const int*   pos      = (const int*)d_in[13];

  float* ws  = (float*)d_ws;
  float* h   = ws;                 // 4096
  float* q   = ws + 4096;          // 4096
  float* k   = ws + 8192;          // 4096
  float* v   = ws + 12288;         // 4096
  float* o   = ws + 16384;         // 4096
  float* x2  = ws + 20480;         // 4096
  float* h2  = ws + 24576;         // 4096
  float* ff1 = ws + 28672;         // 11008
  float* po  = ws + 28672 + 11008; // 32*8*128 = 32768
  float* pm  = po + 32768;         // 256
  float* pl  = pm + 256;           // 256
  float* out = (float*)d_out;

  // --- attention ---
  k_rmsnorm<<<1, 256, 0, stream>>>(x, attn_nrm, h, HIDDEN);
  k_gemv_wmma<<<HIDDEN / 16, 256, HIDDEN * 4, stream>>>(w_q, h, nullptr, q, HIDDEN, HIDDEN, 0);
  k_gemv_wmma<<<HIDDEN / 16, 256, HIDDEN * 4, stream>>>(w_k, h, nullptr, k, HIDDEN, HIDDEN, 0);
  k_gemv_wmma<<<HIDDEN / 16, 256, HIDDEN * 4, stream>>>(w_v, h, nullptr, v, HIDDEN, HIDDEN, 0);
  k_rope<<<8, 256, 0, stream>>>(q, k, sin_c, cos_c, pos);
  k_attn_partial<<<dim3(NHEADS, SPLITS), 256, 0, stream>>>(q, k_cache, v_cache, po, pm, pl);
  k_attn_combine<<<NHEADS, HDIM, 0, stream>>>(q, k, v, po, pm, pl, o);
  k_gemv_wmma<<<HIDDEN / 16, 256, HIDDEN * 4, stream>>>(w_o, o, x, x2, HIDDEN, HIDDEN, 2);
  // --- MLP ---
  k_rmsnorm<<<1, 256, 0, stream>>>(x2, ffn_nrm, h2, HIDDEN);
  k_gemv_wmma<<<INTERM / 16, 256, HIDDEN * 4, stream>>>(w_ff1, h2, nullptr, ff1, HIDDEN, INTERM, 1);
  k_gemv_wmma<<<HIDDEN / 16, 256, INTERM * 4, stream>>>(w_ff2, ff1, x2, out, INTERM, HIDDEN, 2);
}